// GCNMultiInputPredictor_23545010717185
// MI455X (gfx1250) — compile-verified
//
#include <hip/hip_runtime.h>
#include <math.h>

#define N_NODES  100000
#define N_EDGES  1600000
#define N_GRAPHS 1024

typedef __attribute__((ext_vector_type(2))) float v2f;
typedef __attribute__((ext_vector_type(8))) float v8f;

// ---------------------------------------------------------------------------
// fp32 WMMA GEMM:  H[M,256] = X[M,K] @ W[K,256]
// Block = 128 threads = 4 waves; each wave owns a 64-col strip (4 n-tiles)
// and register-blocks TWO m-tiles (32 rows) so each B fragment feeds 2 WMMAs.
// Requires M % 32 == 0, K % 4 == 0; EXEC all-1s (full waves, no branches).
//
// VGPR layouts per CDNA5 ISA 7.12.2:
//  A (16x4 f32):  lane L holds row M=(L&15), K=khalf..khalf+1 (khalf = 2*(L>>4))
//  B (4x16 f32):  lane L holds col N=(L&15), rows K=khalf, khalf+1
//  C/D (16x16):   lane L holds col N=(L&15); VGPR r holds row M = r + 8*(L>>4)
// ---------------------------------------------------------------------------
__global__ __launch_bounds__(128) void gemm_f32_wmma(const float* __restrict__ X,
                                                     const float* __restrict__ W,
                                                     float* __restrict__ H, int K) {
  const int wave  = threadIdx.x >> 5;
  const int lane  = threadIdx.x & 31;
  const int mbase = blockIdx.x * 32;      // two 16-row m-tiles per block
  const int nbase = wave * 64;            // 4 waves cover 256 output cols
  const int l15   = lane & 15;
  const int khalf = (lane >> 4) * 2;      // 0 or 2

  const float* xrow0 = X + (long)(mbase + l15) * K + khalf;
  const float* xrow1 = xrow0 + (long)16 * K;

  v8f acc0[4] = {};   // m-tile 0, n-tiles 0..3
  v8f acc1[4] = {};   // m-tile 1, n-tiles 0..3

  for (int k = 0; k < K; k += 4) {
    v2f a0, a1;
    a0.x = xrow0[k];
    a0.y = xrow0[k + 1];
    a1.x = xrow1[k];
    a1.y = xrow1[k + 1];

    const float* w0 = W + (long)(k + khalf) * 256 + nbase + l15;
    const float* w1 = w0 + 256;

    v2f b[4];
#pragma unroll
    for (int t = 0; t < 4; ++t) {
      b[t].x = w0[t * 16];
      b[t].y = w1[t * 16];
    }
#pragma unroll
    for (int t = 0; t < 4; ++t) {
      acc0[t] = __builtin_amdgcn_wmma_f32_16x16x4_f32(false, a0, false, b[t], (short)0, acc0[t], false, false);
      acc1[t] = __builtin_amdgcn_wmma_f32_16x16x4_f32(false, a1, false, b[t], (short)0, acc1[t], false, false);
    }
  }

  const int mhi = (lane >> 4) * 8;
  float* hrow0 = H + (long)mbase * 256 + nbase + l15;
  float* hrow1 = hrow0 + (long)16 * 256;
#pragma unroll
  for (int r = 0; r < 8; ++r) {
    long off = (long)(r + mhi) * 256;
#pragma unroll
    for (int t = 0; t < 4; ++t) {
      hrow0[off + t * 16] = acc0[t][r];
      hrow1[off + t * 16] = acc1[t][r];
    }
  }
}

// ---------------------------------------------------------------------------
__global__ void fill_f32(float* __restrict__ p, float v, long n) {
  long i = (long)blockIdx.x * blockDim.x + threadIdx.x;
  if (i < n) p[i] = v;
}

__global__ void deg_count(const int* __restrict__ dst, float* __restrict__ deg) {
  int e = blockIdx.x * blockDim.x + threadIdx.x;
  if (e < N_EDGES) atomicAdd(&deg[dst[e]], 1.0f);
}

__global__ void rsqrt_deg(float* __restrict__ deg) {
  int i = blockIdx.x * blockDim.x + threadIdx.x;
  if (i < N_NODES) deg[i] = rsqrtf(deg[i] + 1.0f);
}

// One wave per edge: gather h[src][0:256], scale by dinv[src]*dinv[dst],
// atomically accumulate into agg[dst]. 8 floats per lane (2x float4 loads).
__global__ __launch_bounds__(256) void edge_scatter(const int* __restrict__ src,
                                                    const int* __restrict__ dst,
                                                    const float* __restrict__ dinv,
                                                    const float* __restrict__ h,
                                                    float* __restrict__ agg) {
  const int wave = threadIdx.x >> 5;
  const int lane = threadIdx.x & 31;
  const int e = blockIdx.x * 8 + wave;      // N_EDGES % 8 == 0
  const int s = src[e];
  const int d = dst[e];
  const float norm = dinv[s] * dinv[d];

  const float4* hs = (const float4*)(h + (long)s * 256) + lane * 2;
  float4 h0 = hs[0];
  float4 h1 = hs[1];
  float* ad = agg + (long)d * 256 + lane * 8;
  atomicAdd(ad + 0, h0.x * norm);
  atomicAdd(ad + 1, h0.y * norm);
  atomicAdd(ad + 2, h0.z * norm);
  atomicAdd(ad + 3, h0.w * norm);
  atomicAdd(ad + 4, h1.x * norm);
  atomicAdd(ad + 5, h1.y * norm);
  atomicAdd(ad + 6, h1.z * norm);
  atomicAdd(ad + 7, h1.w * norm);
}

// agg[i,f] += h[i,f]*dinv[i]^2 + b[f]   (thread per float4 chunk)
__global__ void finalize_layer(const float* __restrict__ h,
                               const float* __restrict__ dinv,
                               const float* __restrict__ b,
                               float* __restrict__ agg) {
  const int t = blockIdx.x * blockDim.x + threadIdx.x;  // N_NODES*64 chunks
  const int i = t >> 6;
  const int c = t & 63;
  const float di = dinv[i];
  const float d2 = di * di;
  float4 hv = ((const float4*)h)[t];
  float4 bv = ((const float4*)b)[c];
  float4 av = ((float4*)agg)[t];
  av.x += hv.x * d2 + bv.x;
  av.y += hv.y * d2 + bv.y;
  av.z += hv.z * d2 + bv.z;
  av.w += hv.w * d2 + bv.w;
  ((float4*)agg)[t] = av;
}

// pooled[g][0:256] = 0 (weighted-sum part), pooled[g][256:512] = -inf (max part)
__global__ void fill_pooled(float* __restrict__ pooled) {
  int t = blockIdx.x * blockDim.x + threadIdx.x;  // N_GRAPHS*512
  pooled[t] = ((t & 511) < 256) ? 0.0f : -__builtin_inff();
}

__device__ inline void atomicMaxF(float* addr, float v) {
  if (v >= 0.0f) atomicMax((int*)addr, __float_as_int(v));
  else           atomicMin((unsigned int*)addr, __float_as_uint(v));
}

// One wave per node: gate = sigmoid(feats . w_ws + b_ws) via wave32 reduction,
// then atomic weighted-sum + atomic max into pooled[batch[i]] ([G,512] layout).
__global__ __launch_bounds__(256) void node_pool(const float* __restrict__ feats,
                                                 const float* __restrict__ w_ws,
                                                 const float* __restrict__ b_ws,
                                                 const int* __restrict__ batch,
                                                 float* __restrict__ pooled) {
  const int wave = threadIdx.x >> 5;
  const int lane = threadIdx.x & 31;
  const int i = blockIdx.x * 8 + wave;      // N_NODES % 8 == 0

  const float4* f4 = (const float4*)(feats + (long)i * 256) + lane * 2;
  float4 f0 = f4[0];
  float4 f1 = f4[1];
  const float4* w4 = (const float4*)w_ws + lane * 2;
  float4 w0 = w4[0];
  float4 w1 = w4[1];

  float s = f0.x * w0.x + f0.y * w0.y + f0.z * w0.z + f0.w * w0.w +
            f1.x * w1.x + f1.y * w1.y + f1.z * w1.z + f1.w * w1.w;
#pragma unroll
  for (int m = 16; m >= 1; m >>= 1) s += __shfl_xor(s, m, 32);
  const float gate = 1.0f / (1.0f + __expf(-(s + b_ws[0])));

  const int g = batch[i];
  float* ps = pooled + (long)g * 512 + lane * 8;  // weighted-sum half
  float* pm = ps + 256;                           // max half
  atomicAdd(ps + 0, f0.x * gate); atomicAdd(ps + 1, f0.y * gate);
  atomicAdd(ps + 2, f0.z * gate); atomicAdd(ps + 3, f0.w * gate);
  atomicAdd(ps + 4, f1.x * gate); atomicAdd(ps + 5, f1.y * gate);
  atomicAdd(ps + 6, f1.z * gate); atomicAdd(ps + 7, f1.w * gate);
  atomicMaxF(pm + 0, f0.x); atomicMaxF(pm + 1, f0.y);
  atomicMaxF(pm + 2, f0.z); atomicMaxF(pm + 3, f0.w);
  atomicMaxF(pm + 4, f1.x); atomicMaxF(pm + 5, f1.y);
  atomicMaxF(pm + 6, f1.z); atomicMaxF(pm + 7, f1.w);
}

// out[g] = b_f + sum_k (g2[g,k]+b_go[k])*W_f[k] + sum_j add[g,j]*W_f[256+j]
__global__ void final_out(const float* __restrict__ g2,
                          const float* __restrict__ b_go,
                          const float* __restrict__ add_in,
                          const float* __restrict__ W_f,
                          const float* __restrict__ b_f,
                          float* __restrict__ out) {
  const int g = blockIdx.x * blockDim.x + threadIdx.x;  // N_GRAPHS
  float s = b_f[0];
  for (int k = 0; k < 256; ++k) s += (g2[(long)g * 256 + k] + b_go[k]) * W_f[k];
  for (int j = 0; j < 8; ++j)   s += add_in[g * 8 + j] * W_f[256 + j];
  out[g] = s;
}

// ---------------------------------------------------------------------------
extern "C" void kernel_launch(void* const* d_in, const int* in_sizes, int n_in,
                              void* d_out, int out_size, void* d_ws, size_t ws_size,
                              hipStream_t stream) {
  const float* x     = (const float*)d_in[0];
  const int*   edge  = (const int*)d_in[1];
  const int*   src   = edge;
  const int*   dst   = edge + N_EDGES;
  const int*   batch = (const int*)d_in[2];
  const float* addi  = (const float*)d_in[3];
  const float* W1 = (const float*)d_in[4];  const float* b1 = (const float*)d_in[5];
  const float* W2 = (const float*)d_in[6];  const float* b2 = (const float*)d_in[7];
  const float* W3 = (const float*)d_in[8];  const float* b3 = (const float*)d_in[9];
  const float* w_ws = (const float*)d_in[10]; const float* b_ws = (const float*)d_in[11];
  const float* W_go = (const float*)d_in[12]; const float* b_go = (const float*)d_in[13];
  const float* W_f  = (const float*)d_in[14]; const float* b_f  = (const float*)d_in[15];
  float* out = (float*)d_out;

  char* ws = (char*)d_ws;
  const size_t NB = (size_t)N_NODES * 256 * sizeof(float);   // 102.4 MB
  float* H    = (float*)(ws);               // h = in @ W
  float* A    = (float*)(ws + NB);          // layer out (ping)
  float* B    = (float*)(ws + 2 * NB);      // layer out (pong)
  float* dinv = (float*)(ws + 3 * NB);
  float* pooled = (float*)(ws + 3 * NB + (((size_t)N_NODES * sizeof(float) + 255ul) & ~255ul));
  float* g2 = pooled + (size_t)N_GRAPHS * 512;

  const long NF = (long)N_NODES * 256;

  // degrees -> dinv (in place)
  fill_f32<<<(N_NODES + 255) / 256, 256, 0, stream>>>(dinv, 0.0f, N_NODES);
  deg_count<<<(N_EDGES + 255) / 256, 256, 0, stream>>>(dst, dinv);
  rsqrt_deg<<<(N_NODES + 255) / 256, 256, 0, stream>>>(dinv);

  // layer 1: x[N,128] -> A
  gemm_f32_wmma<<<N_NODES / 32, 128, 0, stream>>>(x, W1, H, 128);
  fill_f32<<<(int)(NF / 256), 256, 0, stream>>>(A, 0.0f, NF);
  edge_scatter<<<N_EDGES / 8, 256, 0, stream>>>(src, dst, dinv, H, A);
  finalize_layer<<<N_NODES * 64 / 256, 256, 0, stream>>>(H, dinv, b1, A);

  // layer 2: A -> B
  gemm_f32_wmma<<<N_NODES / 32, 128, 0, stream>>>(A, W2, H, 256);
  fill_f32<<<(int)(NF / 256), 256, 0, stream>>>(B, 0.0f, NF);
  edge_scatter<<<N_EDGES / 8, 256, 0, stream>>>(src, dst, dinv, H, B);
  finalize_layer<<<N_NODES * 64 / 256, 256, 0, stream>>>(H, dinv, b2, B);

  // layer 3: B -> A
  gemm_f32_wmma<<<N_NODES / 32, 128, 0, stream>>>(B, W3, H, 256);
  fill_f32<<<(int)(NF / 256), 256, 0, stream>>>(A, 0.0f, NF);
  edge_scatter<<<N_EDGES / 8, 256, 0, stream>>>(src, dst, dinv, H, A);
  finalize_layer<<<N_NODES * 64 / 256, 256, 0, stream>>>(H, dinv, b3, A);

  // pooling: pooled[G,512] = [weighted-sum | max]
  fill_pooled<<<N_GRAPHS * 512 / 256, 256, 0, stream>>>(pooled);
  node_pool<<<N_NODES / 8, 256, 0, stream>>>(A, w_ws, b_ws, batch, pooled);

  // graph GEMM (reuse WMMA kernel): g2[G,256] = pooled[G,512] @ W_go
  gemm_f32_wmma<<<N_GRAPHS / 32, 128, 0, stream>>>(pooled, W_go, g2, 512);

  // final readout
  final_out<<<N_GRAPHS / 256, 256, 0, stream>>>(g2, b_go, addi, W_f, b_f, out);
}